// Conv_89713276879316
// MI455X (gfx1250) — compile-verified
//
#include <hip/hip_runtime.h>

typedef __attribute__((ext_vector_type(2))) float v2f;
typedef __attribute__((ext_vector_type(8))) float v8f;

#define KS 15
#define PAD 7
#define IMG 512
#define REG 64                         // 64x64 output region per block
#define IN_ROWS 78                     // REG + KS - 1
#define IN_COLS 80                     // padded width (even stride, k=31 slack)
#define IN_ELEMS (IN_ROWS * IN_COLS)   // 6240 floats
#define BT_ELEMS (KS * 16 * 32)        // 7680 floats: B_t[ky][x][k]

__global__ __launch_bounds__(256, 4)   // cap VGPRs <= 256: no MSB switching, 4 waves/SIMD
void conv15_wmma(const float* __restrict__ img,
                 const float* __restrict__ kern,
                 float* __restrict__ out)
{
    __shared__ float lds[IN_ELEMS + BT_ELEMS];
    float* inl = lds;
    float* bt  = lds + IN_ELEMS;

    const int tid = threadIdx.x;
    const int b   = blockIdx.z;
    const int gx0 = blockIdx.x * REG;
    const int gy0 = blockIdx.y * REG;

    // ---- cooperative load: channel 2 only, with zero halo ----
    const float* src = img + ((size_t)b * 3 + 2) * (size_t)IMG * IMG;
    for (int idx = tid; idx < IN_ELEMS; idx += 256) {
        int r  = idx / IN_COLS;
        int c  = idx - r * IN_COLS;
        int gr = gy0 - PAD + r;
        int gc = gx0 - PAD + c;
        float v = 0.f;
        if ((unsigned)gr < (unsigned)IMG && (unsigned)gc < (unsigned)IMG)
            v = src[(size_t)gr * IMG + gc];
        inl[idx] = v;
    }

    // ---- build banded Toeplitz weight table B_t[ky][x][k] = w[ky][k-x] ----
    for (int idx = tid; idx < BT_ELEMS; idx += 256) {
        int ky  = idx >> 9;        // / 512
        int rem = idx & 511;
        int x   = rem >> 5;        // / 32
        int k   = rem & 31;
        int kx  = k - x;
        bt[idx] = (kx >= 0 && kx < KS) ? kern[ky * KS + kx] : 0.f;
    }

    __syncthreads();

    const int wave = tid >> 5;
    const int lane = tid & 31;
    const int half = lane >> 4;   // 0: lanes 0-15, 1: lanes 16-31
    const int l16  = lane & 15;

    // wave w owns two horizontally adjacent 16x16 tiles (same row band ->
    // identical Toeplitz B operand, shared across both accumulation chains)
    const int lty  = (wave >> 1) * 16;
    const int ltx0 = (wave & 1) * 32;

    v8f accA = {};   // tile at ltx0
    v8f accB = {};   // tile at ltx0 + 16
    for (int ky = 0; ky < KS; ++ky) {
        // A (16x4 f32 layout): lane holds M=l16; K = kk + 2*half + {0,1}
        const float* arow = inl + (lty + l16 + ky) * IN_COLS + ltx0 + 2 * half;
        // B (4x16 f32 layout): lane holds N=l16; K = kk + 2*half + {0,1}
        const float* brow = bt + ((ky * 16 + l16) << 5) + 2 * half;
        #pragma unroll
        for (int c8 = 0; c8 < 8; ++c8) {
            v2f bb, a0, a1;
            bb.x = brow[c8 * 4];
            bb.y = brow[c8 * 4 + 1];
            a0.x = arow[c8 * 4];
            a0.y = arow[c8 * 4 + 1];
            a1.x = arow[16 + c8 * 4];
            a1.y = arow[16 + c8 * 4 + 1];
            accA = __builtin_amdgcn_wmma_f32_16x16x4_f32(
                false, a0, false, bb, (short)0, accA, false, false);
            accB = __builtin_amdgcn_wmma_f32_16x16x4_f32(
                false, a1, false, bb, (short)0, accB, false, false);
        }
    }

    // C/D layout: VGPR r -> row (r + 8*half), col = l16; broadcast to 3 channels
    float* outb = out + (size_t)b * 3 * IMG * IMG;
    const int colA = gx0 + ltx0 + l16;
    #pragma unroll
    for (int r = 0; r < 8; ++r) {
        int row  = gy0 + lty + r + 8 * half;
        size_t o = (size_t)row * IMG + colA;
        float vA = accA[r];
        float vB = accB[r];
        outb[o]                              = vA;
        outb[o + 16]                         = vB;
        outb[o + (size_t)IMG * IMG]          = vA;
        outb[o + (size_t)IMG * IMG + 16]     = vB;
        outb[o + 2 * (size_t)IMG * IMG]      = vA;
        outb[o + 2 * (size_t)IMG * IMG + 16] = vB;
    }
}

extern "C" void kernel_launch(void* const* d_in, const int* in_sizes, int n_in,
                              void* d_out, int out_size, void* d_ws, size_t ws_size,
                              hipStream_t stream) {
    const float* img  = (const float*)d_in[0];
    const float* kern = (const float*)d_in[1];
    float* out        = (float*)d_out;

    dim3 grid(IMG / REG, IMG / REG, 32);   // 8 x 8 x 32 = 2048 blocks
    dim3 block(256);                        // 8 waves (wave32)
    conv15_wmma<<<grid, block, 0, stream>>>(img, kern, out);
}